// GRU_51127290691960
// MI455X (gfx1250) — compile-verified
//
#include <hip/hip_runtime.h>

typedef __attribute__((ext_vector_type(2))) float v2f;
typedef __attribute__((ext_vector_type(8))) float v8f;

#define T_LEN   262144
#define IN_SZ   64
#define HID     16
#define G3      48   // 3*HID

// ---------------------------------------------------------------------------
// Kernel A: xi[t][g] = sum_k x[t][k]*W_ih[g][k] + b_ih[g] + (g<32 ? b_hh[g] : 0)
// One wave (32 lanes) per 16-row tile; 3 N-blocks x 16 K-chunks of
// V_WMMA_F32_16X16X4_F32 (f32 in, f32 acc — exact fp32 GEMM).
// ---------------------------------------------------------------------------
__global__ __launch_bounds__(256) void xi_gemm_wmma(
    const float* __restrict__ x, const float* __restrict__ W_ih,
    const float* __restrict__ b_ih, const float* __restrict__ b_hh,
    float* __restrict__ xi)
{
    const int lane  = threadIdx.x & 31;
    const int wave  = threadIdx.x >> 5;
    const int tile  = blockIdx.x * 8 + wave;      // 16 timesteps per tile
    const int m     = lane & 15;                  // M row / N col within block
    const int khalf = lane >> 4;                  // 0: K={0,1}, 1: K={2,3}

    // A tile: 16x64 of x, 16 chunks of 16x4.  a[i] = x[row m][4i + 2*khalf .. +1]
    const float* xrow = x + (size_t)(tile * 16 + m) * IN_SZ + 2 * khalf;
    v2f a[16];
#pragma unroll
    for (int i = 0; i < 16; ++i)
        a[i] = *(const v2f*)(xrow + 4 * i);

    for (int nb = 0; nb < 3; ++nb) {
        const int col  = nb * 16 + m;             // output gate column (N)
        const float bias = b_ih[col] + (nb < 2 ? b_hh[col] : 0.0f);
        v8f c = {bias, bias, bias, bias, bias, bias, bias, bias};

        // B block: B[k][n] = W_ih[col][k];  b = W_ih[col][4i + 2*khalf .. +1]
        const float* wrow = W_ih + (size_t)col * IN_SZ + 2 * khalf;
#pragma unroll
        for (int i = 0; i < 16; ++i) {
            v2f b = *(const v2f*)(wrow + 4 * i);
            c = __builtin_amdgcn_wmma_f32_16x16x4_f32(
                    false, a[i], false, b, (short)0, c, false, false);
        }

        // D: VGPR r -> row (r + 8*khalf), col = nb*16 + (lane&15)
        float* orow = xi + (size_t)(tile * 16 + 8 * khalf) * G3 + col;
#pragma unroll
        for (int r = 0; r < 8; ++r)
            orow[(size_t)r * G3] = c[r];
    }
}

// ---------------------------------------------------------------------------
// Kernel B: single-wave sequential GRU scan with fused (fc2 o fc1) head.
// lane j (=lane&15) owns gate rows j, 16+j, 32+j of W_hh (48 VGPRs of weights).
// h replicated across lanes via LDS (1 ds_store + 4 ds_load_b128 per step).
// ---------------------------------------------------------------------------
__global__ __launch_bounds__(32) void gru_scan(
    const float* __restrict__ xi, const float* __restrict__ W_hh,
    const float* __restrict__ b_hh, const float* __restrict__ fc1_w,
    const float* __restrict__ fc1_b, const float* __restrict__ fc2_w,
    const float* __restrict__ fc2_b, float* __restrict__ out)
{
    const int lane = threadIdx.x;
    const int j    = lane & 15;

    // Per-lane recurrent weight rows.
    float wr[HID], wz[HID], wn[HID];
#pragma unroll
    for (int k = 0; k < HID; ++k) {
        wr[k] = W_hh[(j)      * HID + k];
        wz[k] = W_hh[(16 + j) * HID + k];
        wn[k] = W_hh[(32 + j) * HID + k];
    }
    const float bhn = b_hh[32 + j];

    // Folded head: w_eff[j] = sum_i fc2_w[i]*fc1_w[i][j];  b_eff scalar.
    float weff = 0.0f, beff = fc2_b[0];
#pragma unroll
    for (int i = 0; i < 8; ++i) {
        const float f2 = fc2_w[i];
        weff += f2 * fc1_w[i * HID + j];
        beff += f2 * fc1_b[i];
    }

    __shared__ __align__(16) float hsh[HID];

    float4 ha = {0,0,0,0}, hb = {0,0,0,0}, hc = {0,0,0,0}, hd = {0,0,0,0};
    float  hj = 0.0f;   // this lane's h[j]

    const float* xr_p = xi + j;
    const float* xz_p = xi + 16 + j;
    const float* xn_p = xi + 32 + j;
    float xr = xr_p[0], xz = xz_p[0], xn = xn_p[0];

    for (int t = 0; t < T_LEN; ++t) {
        // Prefetch next timestep's row (pad row t==T_LEN-1 reads scratch pad).
        const size_t nofs = (size_t)(t + 1) * G3;
        const float xr_n = xr_p[nofs], xz_n = xz_p[nofs], xn_n = xn_p[nofs];

        // hi = W_hh(row j / 16+j / 32+j) . h    (fully unrolled, regs only)
        float hr = 0.0f, hz = 0.0f, hn = bhn;
        const float hv[HID] = {ha.x, ha.y, ha.z, ha.w,  hb.x, hb.y, hb.z, hb.w,
                               hc.x, hc.y, hc.z, hc.w,  hd.x, hd.y, hd.z, hd.w};
#pragma unroll
        for (int k = 0; k < HID; ++k) {
            hr += wr[k] * hv[k];
            hz += wz[k] * hv[k];
            hn += wn[k] * hv[k];
        }

        // Gates: branch-free sigmoid/tanh via v_exp + v_rcp.
        const float r = __builtin_amdgcn_rcpf(1.0f + __expf(-(xr + hr)));
        const float z = __builtin_amdgcn_rcpf(1.0f + __expf(-(xz + hz)));
        const float u = xn + r * hn;
        const float n = 1.0f - 2.0f * __builtin_amdgcn_rcpf(__expf(2.0f * u) + 1.0f);
        const float hnew = n + z * (hj - n);
        hj = hnew;

        // Fused head: y[t] = sum_j hnew_j * weff_j + beff  (16-lane xor-reduce)
        float yv = hnew * weff;
        yv += __shfl_xor(yv, 1, 32);
        yv += __shfl_xor(yv, 2, 32);
        yv += __shfl_xor(yv, 4, 32);
        yv += __shfl_xor(yv, 8, 32);
        if (lane == 0) out[t] = yv + beff;

        // Broadcast h_new to all lanes for next step.
        if (lane < 16) hsh[lane] = hnew;
        __syncthreads();
        ha = *(const float4*)(hsh + 0);
        hb = *(const float4*)(hsh + 4);
        hc = *(const float4*)(hsh + 8);
        hd = *(const float4*)(hsh + 12);
        __syncthreads();

        xr = xr_n; xz = xz_n; xn = xn_n;
    }
}

// ---------------------------------------------------------------------------
extern "C" void kernel_launch(void* const* d_in, const int* in_sizes, int n_in,
                              void* d_out, int out_size, void* d_ws, size_t ws_size,
                              hipStream_t stream) {
    const float* x     = (const float*)d_in[0];
    const float* W_ih  = (const float*)d_in[1];
    const float* W_hh  = (const float*)d_in[2];
    const float* b_ih  = (const float*)d_in[3];
    const float* b_hh  = (const float*)d_in[4];
    const float* fc1_w = (const float*)d_in[5];
    const float* fc1_b = (const float*)d_in[6];
    const float* fc2_w = (const float*)d_in[7];
    const float* fc2_b = (const float*)d_in[8];
    float* out = (float*)d_out;

    float* xi = (float*)d_ws;   // (T_LEN + 1 pad row) * 48 floats ~ 50.3 MB

    // Phase 1: WMMA GEMM, 16384 tiles, 8 waves/block.
    xi_gemm_wmma<<<T_LEN / 16 / 8, 256, 0, stream>>>(x, W_ih, b_ih, b_hh, xi);

    // Phase 2+3: sequential scan with fused head, one wave.
    gru_scan<<<1, 32, 0, stream>>>(xi, W_hh, b_hh, fc1_w, fc1_b, fc2_w, fc2_b, out);
}